// CavityLoss_43722767073667
// MI455X (gfx1250) — compile-verified
//
#include <hip/hip_runtime.h>

// ---------------------------------------------------------------------------
// loss = ( sum over voxels where gt==1 && pred*255 < 128 of -log(pred) ) / 192^3
//   - 5^3 dilation cancels algebraically (center voxel always counted)
//   - non-critical BCE term is -log(1 - 1e-12) == -log(1.0f) == 0 in fp32
// Memory-bound: 56.6 MB streamed once -> ~2.4 us at 23.3 TB/s. WMMA f32
// 16x16x4 (exact fp32) performs the 64->16 cross-lane reduction; DPP row ops
// finish 16->1 with no LDS traffic and no ds-counter waits.
// ---------------------------------------------------------------------------

#define D_SIDE    192
#define N_VOX     (D_SIDE * D_SIDE * D_SIDE)          // 7,077,888
#define N_VEC     (N_VOX / 4)                          // 1,769,472 float4
#define BLOCKS_A  768
#define THREADS_A 256
#define ITERS_A   (N_VEC / (BLOCKS_A * THREADS_A))     // == 9 exactly

typedef float f4  __attribute__((ext_vector_type(4)));
typedef float v2f __attribute__((ext_vector_type(2)));
typedef float v8f __attribute__((ext_vector_type(8)));

// One v_add_f32 whose second operand comes through DPP16 (no LDS, no waits).
#define DPP_ADD(x, ctrl)                                                      \
    ((x) + __builtin_bit_cast(float, __builtin_amdgcn_update_dpp(             \
               0, __builtin_bit_cast(int, (x)), (ctrl), 0xF, 0xF, true)))

#define DPP_QUAD_XOR1 0xB1   // quad_perm [1,0,3,2]
#define DPP_QUAD_XOR2 0x4E   // quad_perm [2,3,0,1]
#define DPP_ROW_ROR4  0x124  // rotate row of 16 by 4 (quads are uniform here)
#define DPP_ROW_ROR8  0x128  // rotate row of 16 by 8

// Reduce 64 per-lane partials (2 accumulators x 32 lanes) to one wave sum.
// WMMA: A = ones(16x4), B VGPR0/1 = acc0/acc1 (rows K0,K2 / K1,K3) =>
//   D[m,n] = acc0[n] + acc1[n] + acc0[n+16] + acc1[n+16]   (exact fp32 colsum)
// Lanes 16-31 hold mirrored rows, so only 4 row-local DPP adds remain.
// Result valid in lane 0 (in fact in every lane of each 16-lane row).
__device__ __forceinline__ float wave_reduce64(float acc0, float acc1) {
#if __has_builtin(__builtin_amdgcn_wmma_f32_16x16x4_f32)
    v2f a; a.x = 1.0f; a.y = 1.0f;
    v2f b; b.x = acc0; b.y = acc1;
    v8f c = {};
    c = __builtin_amdgcn_wmma_f32_16x16x4_f32(
        /*neg_a=*/false, a, /*neg_b=*/false, b,
        /*c_mod=*/(short)0, c, /*reuse_a=*/false, /*reuse_b=*/false);
    float h = c[0];
#else
    float h = acc0 + acc1;
    h += __shfl_xor(h, 16, 32);       // fallback only: fold halves
#endif
    h = DPP_ADD(h, DPP_QUAD_XOR1);    // pair sums within quads
    h = DPP_ADD(h, DPP_QUAD_XOR2);    // quad sums
    h = DPP_ADD(h, DPP_ROW_ROR4);     // + next quad's (uniform) value
    h = DPP_ADD(h, DPP_ROW_ROR8);     // + opposite pair -> row total
    return h;
}

__global__ __launch_bounds__(THREADS_A)
void cavity_partial_kernel(const float* __restrict__ pred,
                           const float* __restrict__ gt,
                           float* __restrict__ partial) {
    const f4* p4 = (const f4*)pred;
    const f4* g4 = (const f4*)gt;
    const int tid  = (int)threadIdx.x;
    const int base = (int)blockIdx.x * THREADS_A + tid;   // fits in 32 bits

    float acc0 = 0.0f, acc1 = 0.0f;   // dual accumulators feed both B VGPRs
#pragma unroll
    for (int i = 0; i < ITERS_A; ++i) {
        const int idx = base + i * (BLOCKS_A * THREADS_A);
        // Single-use streaming data: non-temporal b128 loads keep L2 clean.
        f4 p = __builtin_nontemporal_load(&p4[idx]);
        f4 g = __builtin_nontemporal_load(&g4[idx]);
        // critical = gt==1 && floor(pred*255) < 128  (== pred*255 < 128)
        acc0 += ((g[0] != 0.0f) && (p[0] * 255.0f < 128.0f)) ? -__logf(p[0]) : 0.0f;
        acc0 += ((g[1] != 0.0f) && (p[1] * 255.0f < 128.0f)) ? -__logf(p[1]) : 0.0f;
        acc1 += ((g[2] != 0.0f) && (p[2] * 255.0f < 128.0f)) ? -__logf(p[2]) : 0.0f;
        acc1 += ((g[3] != 0.0f) && (p[3] * 255.0f < 128.0f)) ? -__logf(p[3]) : 0.0f;
    }

    // Uniform trip counts, converged control flow -> EXEC all-ones (WMMA legal).
    const float wsum = wave_reduce64(acc0, acc1);

    __shared__ float lds[THREADS_A / 32];
    const int lane = tid & 31, wv = tid >> 5;
    if (lane == 0) lds[wv] = wsum;
    __syncthreads();
    if (tid == 0) {
        float b = 0.0f;
#pragma unroll
        for (int w = 0; w < THREADS_A / 32; ++w) b += lds[w];
        partial[blockIdx.x] = b;
    }
}

__global__ __launch_bounds__(256)
void cavity_final_kernel(const float* __restrict__ partial,
                         float* __restrict__ out) {
    const int tid = (int)threadIdx.x;
    // 768 partials / 256 threads == 3 each: no divergence, EXEC stays all-ones.
    const float s0 = partial[tid];
    const float s1 = partial[tid + 256];
    const float s2 = partial[tid + 512];

    const float wsum = wave_reduce64(s0 + s1, s2);

    __shared__ float lds[8];
    const int lane = tid & 31, wv = tid >> 5;
    if (lane == 0) lds[wv] = wsum;
    __syncthreads();
    if (tid == 0) {
        float t = 0.0f;
#pragma unroll
        for (int w = 0; w < 8; ++w) t += lds[w];
        out[0] = t * (1.0f / (float)N_VOX);   // BCE mean over all voxels
    }
}

extern "C" void kernel_launch(void* const* d_in, const int* in_sizes, int n_in,
                              void* d_out, int out_size, void* d_ws, size_t ws_size,
                              hipStream_t stream) {
    (void)in_sizes; (void)n_in; (void)out_size; (void)ws_size;
    const float* pred = (const float*)d_in[0];
    const float* gt   = (const float*)d_in[1];
    float* out        = (float*)d_out;
    float* partial    = (float*)d_ws;     // BLOCKS_A floats of scratch

    cavity_partial_kernel<<<dim3(BLOCKS_A), dim3(THREADS_A), 0, stream>>>(pred, gt, partial);
    cavity_final_kernel<<<dim3(1), dim3(256), 0, stream>>>(partial, out);
}